// Aggregator_52905407152978
// MI455X (gfx1250) — compile-verified
//
#include <hip/hip_runtime.h>

typedef __attribute__((ext_vector_type(2))) float v2f;
typedef __attribute__((ext_vector_type(8))) float v8f;

// Problem shape (fixed by the reference).
static constexpr int kN = 16384;
static constexpr int kK = 32;
static constexpr int kD = 256;

// One wave (32 lanes) owns one row n. Block = 256 threads = 8 waves = 8 rows.
// out[n,d] = relu(curr_emb[n,0,d] + sum_k alpha[n,k]*msg[n,k,d])
// computed with V_WMMA_F32_16X16X4_F32:
//   A (16x4 f32)  : alpha chunk broadcast into all 16 rows
//   B (4x16  f32) : msg tile, 16 consecutive d's per WMMA
//   8 chained WMMAs cover K=32; 16 d-tiles cover D=256.
__global__ __launch_bounds__(256) void Aggregator_52905407152978_kernel(
    const float* __restrict__ curr_emb,   // [N, K, D] (only k=0 slice used)
    const float* __restrict__ alpha,      // [N, K, 1]
    const float* __restrict__ msg,        // [N, K, D]
    float* __restrict__ out)              // [N, D]
{
  const int lane = threadIdx.x & 31;
  const int wave = threadIdx.x >> 5;
  const int n    = blockIdx.x * 8 + wave;

  const int hi   = lane >> 4;   // 0: lanes 0-15, 1: lanes 16-31
  const int lcol = lane & 15;

  const float* __restrict__ msg_n = msg      + (size_t)n * (kK * kD);
  const float* __restrict__ cur_n = curr_emb + (size_t)n * (kK * kD); // k=0 row
  const float* __restrict__ alp_n = alpha    + (size_t)n * kK;
  float* __restrict__ out_n       = out      + (size_t)n * kD;

  // A fragments for the 8 K-chunks. ISA layout for f32 16x4 A:
  //   VGPR0: lanes0-15 = K0, lanes16-31 = K2 ; VGPR1: K1 / K3 ; row M = lane&15.
  // All 16 rows identical (alpha broadcast) -> value depends only on half-wave.
  v2f A[8];
#pragma unroll
  for (int c = 0; c < 8; ++c) {
    const int k0 = 4 * c + 2 * hi;
    A[c].x = alp_n[k0];
    A[c].y = alp_n[k0 + 1];
  }

#pragma unroll
  for (int t2 = 0; t2 < 8; ++t2) {            // d-tile pairs: tiles 2*t2, 2*t2+1
    const int d_even = 32 * t2 + lcol;        // columns for even tile (64B run)
    const int d_odd  = d_even + 16;           // odd tile: other half of the line

    v8f acc_e = {};
    v8f acc_o = {};

#pragma unroll
    for (int c = 0; c < 8; ++c) {
      const int krow = 4 * c + 2 * hi;        // B layout mirrors A: K0/K2, K1/K3
      v2f Be, Bo;
      // msg is read exactly once -> non-temporal (don't pollute 192MB L2).
      Be.x = __builtin_nontemporal_load(&msg_n[(size_t)krow       * kD + d_even]);
      Be.y = __builtin_nontemporal_load(&msg_n[(size_t)(krow + 1) * kD + d_even]);
      Bo.x = __builtin_nontemporal_load(&msg_n[(size_t)krow       * kD + d_odd]);
      Bo.y = __builtin_nontemporal_load(&msg_n[(size_t)(krow + 1) * kD + d_odd]);

      acc_e = __builtin_amdgcn_wmma_f32_16x16x4_f32(
          false, A[c], false, Be, (short)0, acc_e, false, false);
      acc_o = __builtin_amdgcn_wmma_f32_16x16x4_f32(
          false, A[c], false, Bo, (short)0, acc_o, false, false);
    }

    // C/D layout: VGPR0, lane l -> row (l<16 ? 0 : 8), col l&15. All A-rows are
    // identical, so VGPR0 carries the tile result in every lane:
    //   lanes 0-15  : even tile, d = 32*t2 + lane
    //   lanes 16-31 : odd  tile, d = 32*t2 + 16 + (lane-16) = 32*t2 + lane
    // -> a single select yields a fully coalesced 128B store.
    const float r  = (hi == 0) ? acc_e[0] : acc_o[0];
    const float cv = __builtin_nontemporal_load(&cur_n[32 * t2 + lane]);
    const float s  = r + cv;
    __builtin_nontemporal_store(s > 0.0f ? s : 0.0f, &out_n[32 * t2 + lane]);
  }
}

extern "C" void kernel_launch(void* const* d_in, const int* in_sizes, int n_in,
                              void* d_out, int out_size, void* d_ws, size_t ws_size,
                              hipStream_t stream) {
  const float* curr_emb = (const float*)d_in[0];
  const float* alpha    = (const float*)d_in[1];
  const float* msg      = (const float*)d_in[2];
  float* out            = (float*)d_out;

  dim3 grid(kN / 8);   // 2048 blocks, 8 rows (waves) each
  dim3 block(256);
  Aggregator_52905407152978_kernel<<<grid, block, 0, stream>>>(curr_emb, alpha,
                                                               msg, out);
}